// SAGE_51677046505876
// MI455X (gfx1250) — compile-verified
//
#include <hip/hip_runtime.h>
#include <hip/hip_bf16.h>

#if defined(__HIP_DEVICE_COMPILE__)
#if !__has_builtin(__builtin_amdgcn_wmma_f32_16x16x4_f32)
#error "missing __builtin_amdgcn_wmma_f32_16x16x4_f32 for gfx1250"
#endif
#endif

typedef __attribute__((ext_vector_type(2))) float v2f;
typedef __attribute__((ext_vector_type(8))) float v8f;

#define F_IN 128  // both layers gather 128-dim features

// ---------------------------------------------------------------------------
// zero a float buffer
__global__ void sage_zero_kernel(float* __restrict__ p, size_t n) {
    size_t i = (size_t)blockIdx.x * blockDim.x + threadIdx.x;
    if (i < n) p[i] = 0.0f;
}

// ---------------------------------------------------------------------------
// per-destination degree: cnt[col[e]] += 1
__global__ void sage_count_kernel(const int* __restrict__ col,
                                  float* __restrict__ cnt, int nE) {
    int e = blockIdx.x * blockDim.x + threadIdx.x;
    if (e < nE) atomicAdd(&cnt[col[e]], 1.0f);
}

// ---------------------------------------------------------------------------
// scatter-add of 128-dim rows: one wave (32 lanes) per edge, float4 per lane.
// Coalesced 512B load per edge from x_src; atomic f32 adds into L2-resident sum.
__global__ __launch_bounds__(256)
void sage_scatter_kernel(const float* __restrict__ xsrc,
                         const int* __restrict__ row,
                         const int* __restrict__ col,
                         float* __restrict__ sum, int nE) {
    size_t gid = (size_t)blockIdx.x * blockDim.x + threadIdx.x;
    int e = (int)(gid >> 5);
    int lane = (int)(gid & 31);
    if (e >= nE) return;
    int r = row[e];
    int c = col[e];
    const float4 v = *(const float4*)(xsrc + (size_t)r * F_IN + lane * 4);
    float* dst = sum + (size_t)c * F_IN + lane * 4;
    atomicAdd(dst + 0, v.x);
    atomicAdd(dst + 1, v.y);
    atomicAdd(dst + 2, v.z);
    atomicAdd(dst + 3, v.w);
}

// ---------------------------------------------------------------------------
// Fused: out[m, :] = (sum[m,:] / max(cnt[m],1)) @ Wl + xdst[m,:] @ Wr + bias
// One block per 16-row tile of nodes. A-tiles staged in LDS (mean applied
// during staging). Each wave computes one 16x16 N-tile of the output via
// V_WMMA_F32_16X16X4_F32, accumulating both GEMMs into one C tile.
//
// Lane layouts (CDNA5 ISA 7.12.2, wave32):
//   A 16x4 f32:  lane L: M = L&15, VGPR{0,1} = K = {2*(L>>4), 2*(L>>4)+1}
//   B 4x16 f32:  lane L: N = L&15, VGPR{0,1} = K = {2*(L>>4), 2*(L>>4)+1}
//   C/D 16x16:   VGPR r: M = r + 8*(L>>4), N = L&15
template <int F_OUT>
__global__ __launch_bounds__(32 * (F_OUT / 16))
void sage_mean_gemm_kernel(const float* __restrict__ sum,
                           const float* __restrict__ cnt,
                           const float* __restrict__ xdst,
                           const float* __restrict__ Wl,   // [F_IN, F_OUT] row-major
                           const float* __restrict__ Wr,   // [F_IN, F_OUT]
                           const float* __restrict__ bias, // [F_OUT]
                           float* __restrict__ out,        // [n_dst, F_OUT]
                           int n_dst) {
    __shared__ float lds_mean[16 * F_IN];
    __shared__ float lds_xdst[16 * F_IN];

    const int row0 = blockIdx.x * 16;
    const int tid = threadIdx.x;
    const int nthreads = 32 * (F_OUT / 16);

    // stage 16 rows of mean(sum/cnt) and xdst into LDS (float4 granularity)
    for (int i = tid; i < 16 * F_IN / 4; i += nthreads) {
        const int r  = i >> 5;        // F_IN/4 = 32 float4 per row
        const int c4 = i & 31;
        const int grow = row0 + r;
        float4 s  = make_float4(0.f, 0.f, 0.f, 0.f);
        float4 xv = make_float4(0.f, 0.f, 0.f, 0.f);
        float inv = 0.0f;
        if (grow < n_dst) {
            inv = 1.0f / fmaxf(cnt[grow], 1.0f);
            s  = *(const float4*)(sum  + (size_t)grow * F_IN + c4 * 4);
            xv = *(const float4*)(xdst + (size_t)grow * F_IN + c4 * 4);
        }
        ((float4*)lds_mean)[i] = make_float4(s.x * inv, s.y * inv, s.z * inv, s.w * inv);
        ((float4*)lds_xdst)[i] = xv;
    }
    __syncthreads();

    const int lane = tid & 31;
    const int wave = tid >> 5;      // N-tile index
    const int half = lane >> 4;     // 0: K pair {0,1}; 1: K pair {2,3}
    const int mrow = lane & 15;     // A-matrix row within tile
    const int ncol = lane & 15;     // B/C column within tile
    const int n0 = wave * 16;

    v8f acc = {};
    // GEMM 1: mean @ Wl
    for (int k0 = 0; k0 < F_IN; k0 += 4) {
        const int k = k0 + 2 * half;
        v2f a; a.x = lds_mean[mrow * F_IN + k];
               a.y = lds_mean[mrow * F_IN + k + 1];
        v2f b; b.x = Wl[(size_t)k * F_OUT + n0 + ncol];
               b.y = Wl[(size_t)(k + 1) * F_OUT + n0 + ncol];
        acc = __builtin_amdgcn_wmma_f32_16x16x4_f32(
                  false, a, false, b, (short)0, acc, false, false);
    }
    // GEMM 2: xdst @ Wr (accumulate into same C tile)
    for (int k0 = 0; k0 < F_IN; k0 += 4) {
        const int k = k0 + 2 * half;
        v2f a; a.x = lds_xdst[mrow * F_IN + k];
               a.y = lds_xdst[mrow * F_IN + k + 1];
        v2f b; b.x = Wr[(size_t)k * F_OUT + n0 + ncol];
               b.y = Wr[(size_t)(k + 1) * F_OUT + n0 + ncol];
        acc = __builtin_amdgcn_wmma_f32_16x16x4_f32(
                  false, a, false, b, (short)0, acc, false, false);
    }

    const float bv = bias[n0 + ncol];
#pragma unroll
    for (int r = 0; r < 8; ++r) {
        const int grow = row0 + r + 8 * half;
        if (grow < n_dst)
            out[(size_t)grow * F_OUT + n0 + ncol] = acc[r] + bv;
    }
}

// ---------------------------------------------------------------------------
static inline void launch_zero(float* p, size_t n, hipStream_t s) {
    size_t blocks = (n + 255) / 256;
    sage_zero_kernel<<<(unsigned)blocks, 256, 0, s>>>(p, n);
}

extern "C" void kernel_launch(void* const* d_in, const int* in_sizes, int n_in,
                              void* d_out, int out_size, void* d_ws, size_t ws_size,
                              hipStream_t stream) {
    // inputs per setup_inputs() order
    const float* x_s  = (const float*)d_in[0];
    const float* x_t  = (const float*)d_in[1];
    const int*   s2t  = (const int*)d_in[2];   // [2, E] : row=src in S, col=dst in T
    const int*   t2s  = (const int*)d_in[3];   // [2, E] : row=src in T, col=dst in S
    const float* W1s_l = (const float*)d_in[4];
    const float* W1s_r = (const float*)d_in[5];
    const float* b1s   = (const float*)d_in[6];
    const float* W1t_l = (const float*)d_in[7];
    const float* W1t_r = (const float*)d_in[8];
    const float* b1t   = (const float*)d_in[9];
    const float* W2s_l = (const float*)d_in[10];
    const float* W2s_r = (const float*)d_in[11];
    const float* b2s   = (const float*)d_in[12];
    const float* W2t_l = (const float*)d_in[13];
    const float* W2t_r = (const float*)d_in[14];
    const float* b2t   = (const float*)d_in[15];

    const int n_s = in_sizes[0] / F_IN;
    const int n_t = in_sizes[1] / F_IN;
    const int nE  = in_sizes[2] / 2;
    const int OUT = in_sizes[10] / F_IN;   // W2s_l is [128, OUT]
    (void)n_in; (void)out_size; (void)ws_size; (void)OUT;

    const int* s2t_row = s2t;          const int* s2t_col = s2t + nE;
    const int* t2s_row = t2s;          const int* t2s_col = t2s + nE;

    // workspace carve-up
    float* ws = (float*)d_ws;
    const size_t nmax = (size_t)(n_s > n_t ? n_s : n_t);
    float* sum   = ws;                          // nmax * 128
    float* cnt_s = sum + nmax * F_IN;           // n_s
    float* cnt_t = cnt_s + n_s;                 // n_t
    float* h_s   = cnt_t + n_t;                 // n_s * 128
    float* h_t   = h_s + (size_t)n_s * F_IN;    // n_t * 128

    float* z_s = (float*)d_out;                     // [n_s, 64]
    float* z_t = (float*)d_out + (size_t)n_s * 64;  // [n_t, 64]

    const int cntBlocks = (nE + 255) / 256;
    const unsigned scatBlocks = (unsigned)(((size_t)nE * 32 + 255) / 256);

    // degrees (shared by both layers; same edge lists)
    launch_zero(cnt_s, (size_t)n_s, stream);
    launch_zero(cnt_t, (size_t)n_t, stream);
    sage_count_kernel<<<cntBlocks, 256, 0, stream>>>(t2s_col, cnt_s, nE);
    sage_count_kernel<<<cntBlocks, 256, 0, stream>>>(s2t_col, cnt_t, nE);

    // ---- layer 1, S side: gather x_t via t2s -> h_s ----
    launch_zero(sum, nmax * F_IN, stream);
    sage_scatter_kernel<<<scatBlocks, 256, 0, stream>>>(x_t, t2s_row, t2s_col, sum, nE);
    sage_mean_gemm_kernel<128><<<(n_s + 15) / 16, 256, 0, stream>>>(
        sum, cnt_s, x_s, W1s_l, W1s_r, b1s, h_s, n_s);

    // ---- layer 1, T side: gather x_s via s2t -> h_t ----
    launch_zero(sum, nmax * F_IN, stream);
    sage_scatter_kernel<<<scatBlocks, 256, 0, stream>>>(x_s, s2t_row, s2t_col, sum, nE);
    sage_mean_gemm_kernel<128><<<(n_t + 15) / 16, 256, 0, stream>>>(
        sum, cnt_t, x_t, W1t_l, W1t_r, b1t, h_t, n_t);

    // ---- layer 2, S side: gather h_t via t2s -> z_s ----
    launch_zero(sum, nmax * F_IN, stream);
    sage_scatter_kernel<<<scatBlocks, 256, 0, stream>>>(h_t, t2s_row, t2s_col, sum, nE);
    sage_mean_gemm_kernel<64><<<(n_s + 15) / 16, 128, 0, stream>>>(
        sum, cnt_s, h_s, W2s_l, W2s_r, b2s, z_s, n_s);

    // ---- layer 2, T side: gather h_s via s2t -> z_t ----
    launch_zero(sum, nmax * F_IN, stream);
    sage_scatter_kernel<<<scatBlocks, 256, 0, stream>>>(h_s, s2t_row, s2t_col, sum, nE);
    sage_mean_gemm_kernel<64><<<(n_t + 15) / 16, 128, 0, stream>>>(
        sum, cnt_t, h_t, W2t_l, W2t_r, b2t, z_t, n_t);
}